// Quantizer_36112085025019
// MI455X (gfx1250) — compile-verified
//
#include <hip/hip_runtime.h>

// ---------------------------------------------------------------------------
// VQ quantizer for MI455X (gfx1250, wave32).
//   dist = ||x||^2 - 2 x.e + ||e||^2 ; argmin over 512 codes; gather; MSE.
// GEMM part (x @ embed) runs on V_WMMA_F32_16X16X4_F32 (fp32 matrix path --
// argmin is precision-sensitive, and at ~17 GFLOP vs ~130 MB traffic the
// kernel is compute/memory balanced, so fp32 WMMA is the sweet spot).
// Round-1 tuning: dual independent WMMA accumulator chains, B-fragment
// register prefetch (one dscnt wait per tile instead of 8), and an XOR-16
// LDS column swizzle so the two lane halves read disjoint bank sets.
// ---------------------------------------------------------------------------

typedef float v2f __attribute__((ext_vector_type(2)));
typedef float v8f __attribute__((ext_vector_type(8)));

#define DIM   64
#define NE    512
#define NROWS (64 * 64 * 64)          // 262144
#define ROWS_PER_BLOCK 128            // 8 waves x 16 rows
#define CPP   128                     // embed columns staged per LDS pass
#define NPASS (NE / CPP)              // 4
#define TILES_PER_PASS (CPP / 16)     // 8

// Prep: per-code squared norms ||e_j||^2 into workspace; zero the diff scalar.
__global__ __launch_bounds__(256) void vq_prep(const float* __restrict__ embed,
                                               float* __restrict__ enorm,
                                               float* __restrict__ diffout) {
  int j = blockIdx.x * blockDim.x + threadIdx.x;
  if (j == 0) diffout[0] = 0.0f;
  if (j < NE) {
    float s = 0.0f;
    for (int d = 0; d < DIM; ++d) {
      float e = embed[d * NE + j];
      s += e * e;
    }
    enorm[j] = s;
  }
}

__global__ __launch_bounds__(256) void vq_main(const float* __restrict__ x,
                                               const float* __restrict__ embed,
                                               const float* __restrict__ enorm,
                                               float* __restrict__ qout,
                                               float* __restrict__ indout,
                                               float* __restrict__ diffout) {
  // B staging: k-pair interleaved (one v2f load per fragment) + XOR-16 column
  // swizzle on odd k-pair rows (bank-decorrelates the two lane halves).
  __shared__ float sB[DIM * CPP];     // 32 KB
  __shared__ int   sMin[8][16];       // per-wave winning code per row

  const int tid  = threadIdx.x;
  const int wave = tid >> 5;
  const int lane = tid & 31;
  const int half = lane >> 4;         // 0: lanes 0-15, 1: lanes 16-31
  const int l16  = lane & 15;

  const int row0 = blockIdx.x * ROWS_PER_BLOCK + wave * 16;

  // ---- A fragments: 16x64 tile, per ISA layout for 16x4 f32 A-matrix.
  // lane half 0 holds K = 4c+{0,1}, half 1 holds K = 4c+{2,3}, row = lane%16.
  const float* xr = x + (size_t)(row0 + l16) * DIM;
  v2f a[16];
#pragma unroll
  for (int c = 0; c < 16; ++c)
    a[c] = ((const v2f*)xr)[2 * c + half];

  float minv[8];
  int   mini[8];
#pragma unroll
  for (int v = 0; v < 8; ++v) { minv[v] = 3.4e38f; mini[v] = 0; }

  for (int pass = 0; pass < NPASS; ++pass) {
    __syncthreads();
    // Stage embed[:, pass*CPP .. +CPP) into LDS:
    //   sB[(kp*CPP + (n ^ ((kp&1)<<4)))*2 + (d&1)],  kp = d>>1.
    for (int i = tid; i < DIM * CPP; i += 256) {
      int d   = i >> 7;            // 0..63
      int n   = i & (CPP - 1);     // 0..127
      int kp  = d >> 1;
      int nsw = n ^ ((kp & 1) << 4);
      sB[(kp * CPP + nsw) * 2 + (d & 1)] =
          embed[d * NE + pass * CPP + n];
    }
    __syncthreads();

    for (int t = 0; t < TILES_PER_PASS; ++t) {
      // Read kp parity == half for every chunk, so the swizzle is a single
      // per-tile XOR on the column index.
      const int ncol = (t * 16 + l16) ^ (half << 4);

      // Prefetch all 16 B fragments -> loads pipeline, one dscnt wait.
      v2f b[16];
#pragma unroll
      for (int c = 0; c < 16; ++c) {
        const int kp = 2 * c + half;
        b[c] = ((const v2f*)sB)[kp * CPP + ncol];
      }

      // Two independent accumulator chains (halve the WMMA RAW chain).
      v8f acc0 = {0.f, 0.f, 0.f, 0.f, 0.f, 0.f, 0.f, 0.f};
      v8f acc1 = {0.f, 0.f, 0.f, 0.f, 0.f, 0.f, 0.f, 0.f};
#pragma unroll
      for (int c = 0; c < 16; c += 2) {
        acc0 = __builtin_amdgcn_wmma_f32_16x16x4_f32(
            false, a[c],     false, b[c],     (short)0, acc0, false, false);
        acc1 = __builtin_amdgcn_wmma_f32_16x16x4_f32(
            false, a[c + 1], false, b[c + 1], (short)0, acc1, false, false);
      }

      // C layout: lanes 0-15 hold rows v, lanes 16-31 rows v+8; col = lane%16.
      const int gcol = pass * CPP + t * 16 + l16;
      const float en = enorm[gcol];                 // 2 KB table, L2-resident
#pragma unroll
      for (int v = 0; v < 8; ++v) {
        float sc = en - 2.0f * (acc0[v] + acc1[v]); // ||x||^2 omitted: row-const
        if (sc < minv[v]) { minv[v] = sc; mini[v] = gcol; }
      }
    }
  }

  // ---- argmin across the 16 lanes of each half (columns), tie -> lowest idx.
#pragma unroll
  for (int off = 1; off < 16; off <<= 1) {
#pragma unroll
    for (int v = 0; v < 8; ++v) {
      float ov = __shfl_xor(minv[v], off, 32);
      int   oi = __shfl_xor(mini[v], off, 32);
      if (ov < minv[v] || (ov == minv[v] && oi < mini[v])) {
        minv[v] = ov; mini[v] = oi;
      }
    }
  }
  if (l16 == 0) {
#pragma unroll
    for (int v = 0; v < 8; ++v) sMin[wave][v + 8 * half] = mini[v];
  }
  __syncthreads();

  // ---- gather winning codes, write quantize (coalesced f32x2), MSE partials.
  float dsum = 0.0f;
#pragma unroll 4
  for (int m = 0; m < 16; ++m) {
    const int idx = sMin[wave][m];
    const size_t orow = (size_t)(row0 + m) * DIM;
    const int d = lane * 2;
    float q0 = embed[(d    ) * NE + idx];
    float q1 = embed[(d + 1) * NE + idx];
    v2f xv = ((const v2f*)(x + orow))[lane];
    v2f qv; qv.x = q0; qv.y = q1;
    ((v2f*)(qout + orow))[lane] = qv;               // quantize_st == quantize
    float e0 = q0 - xv.x, e1 = q1 - xv.y;
    dsum += e0 * e0 + e1 * e1;
  }
  if (lane < 16) indout[row0 + lane] = (float)sMin[wave][lane];

#pragma unroll
  for (int off = 16; off >= 1; off >>= 1)
    dsum += __shfl_xor(dsum, off, 32);
  if (lane == 0)
    atomicAdd(diffout, dsum * (1.0f / (float)((size_t)NROWS * DIM)));
}

extern "C" void kernel_launch(void* const* d_in, const int* in_sizes, int n_in,
                              void* d_out, int out_size, void* d_ws, size_t ws_size,
                              hipStream_t stream) {
  const float* x     = (const float*)d_in[0];   // [262144, 64] flattened
  const float* embed = (const float*)d_in[1];   // [64, 512]

  float* qout    = (float*)d_out;                       // 16777216
  float* diffout = qout + (size_t)NROWS * DIM;          // 1
  float* indout  = diffout + 1;                         // 262144
  float* enorm   = (float*)d_ws;                        // 512 floats scratch

  vq_prep<<<(NE + 255) / 256, 256, 0, stream>>>(embed, enorm, diffout);
  vq_main<<<NROWS / ROWS_PER_BLOCK, 256, 0, stream>>>(x, embed, enorm,
                                                      qout, indout, diffout);
}